// LocalProcessing_21646635172629
// MI455X (gfx1250) — compile-verified
//
#include <hip/hip_runtime.h>

// LocalProcessing GNN for MI455X (gfx1250).
// Key idea: the all-pairs edge linear layers are separable (edge features are a
// concat of per-i and per-j vectors), so the O(N^2 * 140 * 64) GEMMs collapse to
// O(N * 70 * 64) per-node projections (done with V_WMMA_F32_16X16X4_F32) plus an
// O(N^2 * 64) elementwise pairwise kernel (VALU + v_rcp_f32, L2-resident data).

#define NN   768
#define MAPD 6
#define EMB  64

typedef float v2f __attribute__((ext_vector_type(2)));
typedef float v8f __attribute__((ext_vector_type(8)));

// ---------------------------------------------------------------- weight packing
// WaI rows = Wa{0..5, 12..75}   (the "i" blocks: mi/10, ei)
// WaJ rows = Wa{6..11, 76..139} (the "j" blocks: mj, ej)   ; pad K 70->72 with 0
// Wf1p = Wf1 [65,64] padded to [68,64]
__global__ void pack_weights_kernel(const float* __restrict__ Wa,
                                    const float* __restrict__ We,
                                    const float* __restrict__ Wf1,
                                    float* __restrict__ WaI, float* __restrict__ WaJ,
                                    float* __restrict__ WeI, float* __restrict__ WeJ,
                                    float* __restrict__ Wf1p)
{
    int idx = blockIdx.x * blockDim.x + threadIdx.x;
    if (idx >= 72 * 64) return;
    int k = idx >> 6, c = idx & 63;
    int ri = (k < 6) ? k     : k + 6;    // -> rows 0..5 then 12..75
    int rj = (k < 6) ? k + 6 : k + 70;   // -> rows 6..11 then 76..139
    float aI = (k < 70) ? Wa[ri * 64 + c] : 0.f;
    float aJ = (k < 70) ? Wa[rj * 64 + c] : 0.f;
    float eI = (k < 70) ? We[ri * 64 + c] : 0.f;
    float eJ = (k < 70) ? We[rj * 64 + c] : 0.f;
    WaI[idx] = aI; WaJ[idx] = aJ; WeI[idx] = eI; WeJ[idx] = eJ;
    if (k < 68) Wf1p[k * 64 + c] = (k < 65) ? Wf1[k * 64 + c] : 0.f;
}

// Xi[i] = [m[i]/10, e[i], 0, 0]   Xj[i] = [m[i], e[i], 0, 0]   (K = 72)
__global__ void pack_x_kernel(const float* __restrict__ m, const float* __restrict__ e,
                              float* __restrict__ Xi, float* __restrict__ Xj)
{
    int idx = blockIdx.x * blockDim.x + threadIdx.x;
    if (idx >= NN * 72) return;
    int i = idx / 72, k = idx - i * 72;
    float xi = 0.f, xj = 0.f;
    if (k < 6)       { float mv = m[i * MAPD + k]; xi = mv * 0.1f; xj = mv; }
    else if (k < 70) { float ev = e[i * EMB + (k - 6)]; xi = ev; xj = ev; }
    Xi[idx] = xi; Xj[idx] = xj;
}

// node_in[i] = [m[i] (6), m[i,0]-m[rep,0], m[i,1]-m[rep,1], agg[i] (64)]  (K = 72)
__global__ void pack_node_in_kernel(const float* __restrict__ m, const float* __restrict__ agg,
                                    const int* __restrict__ rep_p, float* __restrict__ Xn)
{
    int idx = blockIdx.x * blockDim.x + threadIdx.x;
    if (idx >= NN * 72) return;
    int i = idx / 72, k = idx - i * 72;
    int rep = rep_p[0];
    float v;
    if (k < 6)       v = m[i * MAPD + k];
    else if (k == 6) v = m[i * MAPD + 0] - m[rep * MAPD + 0];
    else if (k == 7) v = m[i * MAPD + 1] - m[rep * MAPD + 1];
    else             v = agg[i * EMB + (k - 8)];
    Xn[idx] = v;
}

// fin[i] = [e[i] (64), 1.0, 0, 0, 0]  (K = 68)
__global__ void pack_xf_kernel(const float* __restrict__ e, float* __restrict__ Xf)
{
    int idx = blockIdx.x * blockDim.x + threadIdx.x;
    if (idx >= NN * 68) return;
    int i = idx / 68, k = idx - i * 68;
    Xf[idx] = (k < 64) ? e[i * EMB + k] : ((k == 64) ? 1.0f : 0.0f);
}

// ---------------------------------------------------------------- WMMA f32 GEMM
// C[768,64] = act(A[768,K] @ B[K,64] + bias), K % 4 == 0.
// Block = 128 threads (4 waves); each wave owns a 16(M) x 64(N) tile.
// A frag (16x4 f32): lanes 0-15 -> K {0,1}, lanes 16-31 -> K {2,3} (ISA 7.12.2).
// B frag (4x16 f32): mirrored — lane holds column N=lane%16, rows K per half.
__global__ void wmma_gemm_n64(const float* __restrict__ A, const float* __restrict__ B,
                              const float* __restrict__ bias, float* __restrict__ C,
                              int K, int do_relu)
{
    int wave = threadIdx.x >> 5;
    int lane = threadIdx.x & 31;
    int r0   = blockIdx.x * 64 + wave * 16;
    int half = lane >> 4;
    int l    = lane & 15;

    v8f acc0 = {}, acc1 = {}, acc2 = {}, acc3 = {};
    const float* arow = A + (size_t)(r0 + l) * K + 2 * half;

    for (int k0 = 0; k0 < K; k0 += 4) {
        v2f a;
        a.x = arow[k0];
        a.y = arow[k0 + 1];
        const float* brow = B + (size_t)(k0 + 2 * half) * 64 + l;
        v2f b;
        b.x = brow[0];   b.y = brow[64];
        acc0 = __builtin_amdgcn_wmma_f32_16x16x4_f32(false, a, false, b, (short)0, acc0, false, false);
        b.x = brow[16];  b.y = brow[80];
        acc1 = __builtin_amdgcn_wmma_f32_16x16x4_f32(false, a, false, b, (short)0, acc1, false, false);
        b.x = brow[32];  b.y = brow[96];
        acc2 = __builtin_amdgcn_wmma_f32_16x16x4_f32(false, a, false, b, (short)0, acc2, false, false);
        b.x = brow[48];  b.y = brow[112];
        acc3 = __builtin_amdgcn_wmma_f32_16x16x4_f32(false, a, false, b, (short)0, acc3, false, false);
    }

    // C/D layout: VGPR v, lanes 0-15 -> M = v, lanes 16-31 -> M = v + 8; N = lane%16.
    int row = r0 + half * 8;
    #pragma unroll
    for (int v = 0; v < 8; ++v) {
        float vals[4] = { acc0[v], acc1[v], acc2[v], acc3[v] };
        #pragma unroll
        for (int t = 0; t < 4; ++t) {
            int col = t * 16 + l;
            float x = vals[t];
            if (bias) x += bias[col];
            if (do_relu) x = fmaxf(x, 0.f);
            C[(size_t)(row + v) * 64 + col] = x;
        }
    }
}

// ---------------------------------------------------------------- pairwise core
// agg[i,c] = e[i,c] + (sum_j relu(Ei[i,c]+Ej[j,c]) * w) / (sum_j w),
//            w = 1 / (relu(Ai[i,c]+Aj[j,c])^2 + 0.01)
// Block handles 4 i-rows; 256 threads = 64 channels x 4 j-slices.
__global__ void pairwise_kernel(const float* __restrict__ e,
                                const float* __restrict__ Ai, const float* __restrict__ Aj,
                                const float* __restrict__ Ei, const float* __restrict__ Ej,
                                float* __restrict__ agg)
{
    const int TI = 4;
    int i0    = blockIdx.x * TI;
    int t     = threadIdx.x;
    int c     = t & 63;
    int slice = t >> 6;                       // 0..3

    float ai[TI], ei[TI], num[TI], den[TI];
    #pragma unroll
    for (int q = 0; q < TI; ++q) {
        ai[q]  = Ai[(i0 + q) * EMB + c];
        ei[q]  = Ei[(i0 + q) * EMB + c];
        num[q] = 0.f; den[q] = 0.f;
    }

    for (int j = slice; j < NN; j += 4) {
        float aj = Aj[j * EMB + c];
        float ej = Ej[j * EMB + c];
        #pragma unroll
        for (int q = 0; q < TI; ++q) {
            float a = fmaxf(ai[q] + aj, 0.f);
            float w = __builtin_amdgcn_rcpf(fmaf(a, a, 0.01f));
            float p = fmaxf(ei[q] + ej, 0.f);
            num[q]  = fmaf(p, w, num[q]);
            den[q] += w;
        }
    }

    __shared__ float snum[4][TI][64];
    __shared__ float sden[4][TI][64];
    #pragma unroll
    for (int q = 0; q < TI; ++q) { snum[slice][q][c] = num[q]; sden[slice][q][c] = den[q]; }
    __syncthreads();

    if (slice == 0) {
        #pragma unroll
        for (int q = 0; q < TI; ++q) {
            float n = snum[0][q][c] + snum[1][q][c] + snum[2][q][c] + snum[3][q][c];
            float d = sden[0][q][c] + sden[1][q][c] + sden[2][q][c] + sden[3][q][c];
            agg[(i0 + q) * EMB + c] = e[(i0 + q) * EMB + c] + n / d;
        }
    }
}

// out[768,2] = h[768,64] @ W2[64,2] + b2
__global__ void linear_n2_kernel(const float* __restrict__ h, const float* __restrict__ W2,
                                 const float* __restrict__ b2, float* __restrict__ out)
{
    int idx = blockIdx.x * blockDim.x + threadIdx.x;
    if (idx >= NN * 2) return;
    int i = idx >> 1, o = idx & 1;
    float s = b2[o];
    #pragma unroll
    for (int c = 0; c < EMB; ++c) s = fmaf(h[i * EMB + c], W2[c * 2 + o], s);
    out[idx] = s;
}

// out[0:6] = m[rep,:], out[6:70] = e[rep,:]
__global__ void out_rep_kernel(const float* __restrict__ m, const float* __restrict__ e,
                               const int* __restrict__ rep_p, float* __restrict__ out)
{
    int t = threadIdx.x;
    if (t >= 70) return;
    int rep = rep_p[0];
    out[t] = (t < 6) ? m[rep * MAPD + t] : e[rep * EMB + (t - 6)];
}

// ---------------------------------------------------------------- launch
extern "C" void kernel_launch(void* const* d_in, const int* in_sizes, int n_in,
                              void* d_out, int out_size, void* d_ws, size_t ws_size,
                              hipStream_t stream)
{
    (void)in_sizes; (void)n_in; (void)out_size; (void)ws_size;

    const float* m   = (const float*)d_in[0];
    const float* e0  = (const float*)d_in[1];
    const int*   rep = (const int*)  d_in[2];
    const float* Wa  = (const float*)d_in[3];
    const float* ba  = (const float*)d_in[4];
    const float* We  = (const float*)d_in[5];
    const float* be  = (const float*)d_in[6];
    const float* Wn1 = (const float*)d_in[7];
    const float* bn1 = (const float*)d_in[8];
    const float* Wn2 = (const float*)d_in[9];
    const float* bn2 = (const float*)d_in[10];
    const float* Wp1 = (const float*)d_in[11];
    const float* bp1 = (const float*)d_in[12];
    const float* Wp2 = (const float*)d_in[13];
    const float* bp2 = (const float*)d_in[14];
    const float* Wt1 = (const float*)d_in[15];
    const float* bt1 = (const float*)d_in[16];
    const float* Wt2 = (const float*)d_in[17];
    const float* bt2 = (const float*)d_in[18];
    const float* Wo1 = (const float*)d_in[19];
    const float* bo1 = (const float*)d_in[20];
    const float* Wo2 = (const float*)d_in[21];
    const float* bo2 = (const float*)d_in[22];
    const float* Wf1 = (const float*)d_in[23];
    const float* bf1 = (const float*)d_in[24];
    const float* Wf2 = (const float*)d_in[25];
    const float* bf2 = (const float*)d_in[26];

    float* ws  = (float*)d_ws;
    size_t off = 0;
    auto alloc = [&](size_t n) { float* p = ws + off; off += n; return p; };

    float* eA   = alloc(NN * EMB);
    float* eB   = alloc(NN * EMB);
    float* Aiw  = alloc(NN * EMB);
    float* Ajw  = alloc(NN * EMB);
    float* Eiw  = alloc(NN * EMB);
    float* Ejw  = alloc(NN * EMB);
    float* agg  = alloc(NN * EMB);
    float* h    = alloc(NN * EMB);
    float* Xi   = alloc(NN * 72);
    float* Xj   = alloc(NN * 72);
    float* Xn   = alloc(NN * 72);
    float* Xf   = alloc(NN * 68);
    float* WaI  = alloc(72 * 64);
    float* WaJ  = alloc(72 * 64);
    float* WeI  = alloc(72 * 64);
    float* WeJ  = alloc(72 * 64);
    float* Wf1p = alloc(68 * 64);

    float* out = (float*)d_out;
    // output layout: m_rep(6), e_rep(64), pos(1536), tan(1536), nor(1536), fut1(1536), fut2(1536)
    float* o_pos = out + 70;
    float* o_tan = out + 70 + 1536;
    float* o_nor = out + 70 + 2 * 1536;
    float* o_f1  = out + 70 + 3 * 1536;
    float* o_f2  = out + 70 + 4 * 1536;

    pack_weights_kernel<<<18, 256, 0, stream>>>(Wa, We, Wf1, WaI, WaJ, WeI, WeJ, Wf1p);

    const float* ecur = e0;
    float* enext[2] = { eB, eA };
    for (int it = 0; it < 2; ++it) {
        pack_x_kernel<<<216, 256, 0, stream>>>(m, ecur, Xi, Xj);
        wmma_gemm_n64<<<12, 128, 0, stream>>>(Xi, WaI, nullptr, Aiw, 72, 0);
        wmma_gemm_n64<<<12, 128, 0, stream>>>(Xj, WaJ, ba,      Ajw, 72, 0);
        wmma_gemm_n64<<<12, 128, 0, stream>>>(Xi, WeI, nullptr, Eiw, 72, 0);
        wmma_gemm_n64<<<12, 128, 0, stream>>>(Xj, WeJ, be,      Ejw, 72, 0);
        pairwise_kernel<<<NN / 4, 256, 0, stream>>>(ecur, Aiw, Ajw, Eiw, Ejw, agg);
        pack_node_in_kernel<<<216, 256, 0, stream>>>(m, agg, rep, Xn);
        wmma_gemm_n64<<<12, 128, 0, stream>>>(Xn, Wn1, bn1, h, 72, 1);
        wmma_gemm_n64<<<12, 128, 0, stream>>>(h, Wn2, bn2, enext[it], 64, 0);
        ecur = enext[it];
    }

    // heads: pos / tan / nor
    wmma_gemm_n64<<<12, 128, 0, stream>>>(ecur, Wp1, bp1, h, 64, 1);
    linear_n2_kernel<<<6, 256, 0, stream>>>(h, Wp2, bp2, o_pos);
    wmma_gemm_n64<<<12, 128, 0, stream>>>(ecur, Wt1, bt1, h, 64, 1);
    linear_n2_kernel<<<6, 256, 0, stream>>>(h, Wt2, bt2, o_tan);
    wmma_gemm_n64<<<12, 128, 0, stream>>>(ecur, Wo1, bo1, h, 64, 1);
    linear_n2_kernel<<<6, 256, 0, stream>>>(h, Wo2, bo2, o_nor);

    // future nets: fut1 == fut2 (identical net + input) — hidden layer computed once
    pack_xf_kernel<<<204, 256, 0, stream>>>(ecur, Xf);
    wmma_gemm_n64<<<12, 128, 0, stream>>>(Xf, Wf1p, bf1, h, 68, 1);
    linear_n2_kernel<<<6, 256, 0, stream>>>(h, Wf2, bf2, o_f1);
    linear_n2_kernel<<<6, 256, 0, stream>>>(h, Wf2, bf2, o_f2);

    out_rep_kernel<<<1, 128, 0, stream>>>(m, ecur, rep, out);
}